// TemporalMemory_55533927137637
// MI455X (gfx1250) — compile-verified
//
#include <hip/hip_runtime.h>
#include <math.h>

typedef __attribute__((ext_vector_type(2))) float vf2;
typedef __attribute__((ext_vector_type(8))) float vf8;

// ---------------------------------------------------------------------------
// WMMA f32 16x16x4 fragment helpers (wave32).
// A-matrix 16x4 (MxK) layout per ISA 7.12.2:
//   VGPR0: lanes 0-15 -> (M=lane, K=0), lanes 16-31 -> (M=lane-16, K=2)
//   VGPR1: lanes 0-15 -> (M=lane, K=1), lanes 16-31 -> (M=lane-16, K=3)
// Mirrored layout assumed for the B-matrix 4x16 (KxN).
// ---------------------------------------------------------------------------
__device__ __forceinline__ vf2 frag_kcontig(const float* __restrict__ base,
                                            int row0, int ld, int k, int lane) {
  int r  = row0 + (lane & 15);
  int kk = k + ((lane >> 4) << 1);
  const float* p = base + (size_t)r * ld + kk;
  return vf2{p[0], p[1]};
}

__device__ __forceinline__ vf2 frag_ncontig(const float* __restrict__ base,
                                            int n0, int ld, int k, int lane) {
  int n  = n0 + (lane & 15);
  int kk = k + ((lane >> 4) << 1);
  const float* p = base + (size_t)kk * ld + n;
  return vf2{p[0], p[ld]};
}

__device__ __forceinline__ vf8 wmma_f32(vf2 a, vf2 b, vf8 c) {
  // (neg_a, A, neg_b, B, c_mod, C, reuse_a, reuse_b)
  return __builtin_amdgcn_wmma_f32_16x16x4_f32(false, a, false, b,
                                               (short)0, c, false, false);
}

__device__ __forceinline__ float sigmoidf_(float x) {
  return 1.0f / (1.0f + __expf(-x));
}

// ---------------------------------------------------------------------------
// Kernel 1: fused LSTM cell (gate order i,f,g,o).
// gates = pose @ W_ih^T + h @ W_hh^T + b_ih + b_hh.
// Each wave: 16(M) x 16(N-of-H) tile, 4 accumulators (one per gate).
// Block = 128 threads = 4 waves (16 x 64). Grid = (B/16, H/64).
// ---------------------------------------------------------------------------
__global__ void __launch_bounds__(128)
lstm_cell_wmma(const float* __restrict__ pose,
               const float* __restrict__ hprev,
               const float* __restrict__ cprev,
               const float* __restrict__ W_ih,
               const float* __restrict__ W_hh,
               const float* __restrict__ b_ih,
               const float* __restrict__ b_hh,
               float* __restrict__ h_new,
               float* __restrict__ c_new,
               int H) {
  const int lane = threadIdx.x & 31;
  const int wave = threadIdx.x >> 5;
  const int row0 = blockIdx.x * 16;
  const int n0   = blockIdx.y * 64 + wave * 16;

  vf8 acc[4] = {vf8{}, vf8{}, vf8{}, vf8{}};

  for (int k = 0; k < H; k += 4) {
    vf2 a = frag_kcontig(pose, row0, H, k, lane);
#pragma unroll
    for (int g = 0; g < 4; ++g) {
      vf2 b = frag_kcontig(W_ih, g * H + n0, H, k, lane);
      acc[g] = wmma_f32(a, b, acc[g]);
    }
  }
  for (int k = 0; k < H; k += 4) {
    vf2 a = frag_kcontig(hprev, row0, H, k, lane);
#pragma unroll
    for (int g = 0; g < 4; ++g) {
      vf2 b = frag_kcontig(W_hh, g * H + n0, H, k, lane);
      acc[g] = wmma_f32(a, b, acc[g]);
    }
  }

  const int col   = n0 + (lane & 15);
  const int rbase = row0 + ((lane >> 4) << 3);
  const float bi = b_ih[0 * H + col] + b_hh[0 * H + col];
  const float bf = b_ih[1 * H + col] + b_hh[1 * H + col];
  const float bg = b_ih[2 * H + col] + b_hh[2 * H + col];
  const float bo = b_ih[3 * H + col] + b_hh[3 * H + col];

#pragma unroll
  for (int v = 0; v < 8; ++v) {
    const size_t idx = (size_t)(rbase + v) * H + col;
    const float cv = cprev[idx];
    const float ig = sigmoidf_(acc[0][v] + bi);
    const float fg = sigmoidf_(acc[1][v] + bf);
    const float gg = tanhf(acc[2][v] + bg);
    const float og = sigmoidf_(acc[3][v] + bo);
    const float cn = fg * cv + ig * gg;
    c_new[idx] = cn;
    h_new[idx] = og * tanhf(cn);
  }
}

// ---------------------------------------------------------------------------
// Generic GEMM: C[M,N] = A[M,K] @ op(W) + bias.
// TRANSB=true : W is [N, K] row-major, multiply by W^T (torch Linear).
// TRANSB=false: W is [K, N] row-major (plain matmul).
// Block = 128 threads (4 waves, 16x64 tile). Grid = (M/16, N/64).
// ---------------------------------------------------------------------------
template <bool TRANSB>
__global__ void __launch_bounds__(128)
gemm_wmma(const float* __restrict__ A,
          const float* __restrict__ W,
          const float* __restrict__ bias,
          float* __restrict__ C,
          int N, int K) {
  const int lane = threadIdx.x & 31;
  const int wave = threadIdx.x >> 5;
  const int row0 = blockIdx.x * 16;
  const int n0   = blockIdx.y * 64 + wave * 16;

  vf8 acc = vf8{};
  for (int k = 0; k < K; k += 4) {
    vf2 a = frag_kcontig(A, row0, K, k, lane);
    vf2 b = TRANSB ? frag_kcontig(W, n0, K, k, lane)
                   : frag_ncontig(W, n0, N, k, lane);
    acc = wmma_f32(a, b, acc);
  }

  const int col   = n0 + (lane & 15);
  const int rbase = row0 + ((lane >> 4) << 3);
  const float bb = bias ? bias[col] : 0.0f;
#pragma unroll
  for (int v = 0; v < 8; ++v)
    C[(size_t)(rbase + v) * N + col] = acc[v] + bb;
}

// ---------------------------------------------------------------------------
// Kernel 4: single-pass attention using CDNA5's 320KB-per-WGP LDS.
// The batch row's whole FIFO memory slice (64 x 1024 f32 = 256KB) is staged
// into LDS once, then both the score pass and the weighted-sum pass run out
// of LDS -> memory tensor is read from HBM exactly once (~512MB total).
// mem_b row m (m < M-1) = memory[b, m+1, :]; row M-1 = h_new[b, :].
// scores_m = (Qk_b . mem_b,m) / sqrt(H)   (Q.bk term is softmax-invariant)
// ctx_b    = sum_m softmax(scores)_m * mem_b,m
// One block of 256 threads (8 waves) per batch row; M=64, H=1024 fixed.
// ---------------------------------------------------------------------------
#define ATT_M 64
#define ATT_H 1024

__global__ void __launch_bounds__(256, 1)
attention_lds(const float* __restrict__ memory,
              const float* __restrict__ h_new,
              const float* __restrict__ Qk,
              float* __restrict__ ctx) {
  __shared__ float s_mem[ATT_M * ATT_H];   // 256 KB
  __shared__ float s_qk[ATT_H];            // 4 KB
  __shared__ float s_attn[ATT_M];

  const int b    = blockIdx.x;
  const int tid  = threadIdx.x;
  const int lane = tid & 31;
  const int wave = tid >> 5;

  const float* __restrict__ memb = memory + (size_t)b * ATT_M * ATT_H;
  const float* __restrict__ hb   = h_new + (size_t)b * ATT_H;

  // --- stage Qk_b and the shifted memory slice into LDS (float4 traffic) ---
  for (int i = tid; i < ATT_H; i += 256) s_qk[i] = Qk[(size_t)b * ATT_H + i];

  const int QPERROW = ATT_H / 4;  // 256 float4 per row
  for (int idx = tid; idx < ATT_M * QPERROW; idx += 256) {
    const int m = idx >> 8;          // idx / QPERROW
    const int q = idx & 255;         // idx % QPERROW
    const float4* __restrict__ src =
        (m == ATT_M - 1) ? (const float4*)hb
                         : (const float4*)(memb + (size_t)(m + 1) * ATT_H);
    ((float4*)&s_mem[m * ATT_H])[q] = src[q];
  }
  __syncthreads();

  // --- scores: wave w handles rows [8w, 8w+8), reading LDS ---
  const float inv_sqrt = rsqrtf((float)ATT_H);
#pragma unroll
  for (int i = 0; i < 8; ++i) {
    const int m = wave * 8 + i;
    const float* __restrict__ row = &s_mem[m * ATT_H];
    float s = 0.0f;
    for (int hh = lane; hh < ATT_H; hh += 32) s += row[hh] * s_qk[hh];
#pragma unroll
    for (int off = 16; off > 0; off >>= 1) s += __shfl_xor(s, off, 32);
    if (lane == 0) s_attn[m] = s * inv_sqrt;
  }
  __syncthreads();

  // --- softmax over 64 scores, wave 0 only (2 values per lane) ---
  if (wave == 0) {
    float s0 = s_attn[lane];
    float s1 = s_attn[lane + 32];
    float mx = fmaxf(s0, s1);
#pragma unroll
    for (int off = 16; off > 0; off >>= 1)
      mx = fmaxf(mx, __shfl_xor(mx, off, 32));
    float e0 = __expf(s0 - mx);
    float e1 = __expf(s1 - mx);
    float sum = e0 + e1;
#pragma unroll
    for (int off = 16; off > 0; off >>= 1) sum += __shfl_xor(sum, off, 32);
    const float inv = 1.0f / sum;
    s_attn[lane]      = e0 * inv;
    s_attn[lane + 32] = e1 * inv;
  }
  __syncthreads();

  // --- ctx from LDS: thread t owns columns t, t+256, t+512, t+768 ---
#pragma unroll
  for (int h0 = 0; h0 < ATT_H; h0 += 256) {
    const int hh = h0 + tid;
    float a = 0.0f;
#pragma unroll 8
    for (int m = 0; m < ATT_M; ++m) a += s_attn[m] * s_mem[m * ATT_H + hh];
    ctx[(size_t)b * ATT_H + hh] = a;
  }
}

// ---------------------------------------------------------------------------
// Launcher.
// Outputs (concat): attended [B,H] | h_new [B,H] | c_new [B,H].
// Workspace: Qk [B,H] | ctx [B,H]  (16 MB).
// The attended region of d_out doubles as temp storage for Q (dead after Qk).
// ---------------------------------------------------------------------------
extern "C" void kernel_launch(void* const* d_in, const int* in_sizes, int n_in,
                              void* d_out, int out_size, void* d_ws, size_t ws_size,
                              hipStream_t stream) {
  const float* pose   = (const float*)d_in[0];
  const float* query  = (const float*)d_in[1];
  const float* h      = (const float*)d_in[2];
  const float* c      = (const float*)d_in[3];
  const float* memory = (const float*)d_in[4];
  const float* W_ih   = (const float*)d_in[5];
  const float* W_hh   = (const float*)d_in[6];
  const float* b_ih   = (const float*)d_in[7];
  const float* b_hh   = (const float*)d_in[8];
  const float* Wq     = (const float*)d_in[9];
  const float* bq     = (const float*)d_in[10];
  const float* Wk     = (const float*)d_in[11];
  // d_in[12] = bk: provably unused (adds a per-row constant to scores,
  // which softmax is invariant to).
  const float* Wv     = (const float*)d_in[13];
  const float* bv     = (const float*)d_in[14];

  const int H  = in_sizes[10];             // bq has H elements
  const int B  = in_sizes[0] / H;          // pose_embed is [B, H]

  float* out_att = (float*)d_out;
  float* out_h   = out_att + (size_t)B * H;
  float* out_c   = out_h + (size_t)B * H;
  float* ws_qk   = (float*)d_ws;
  float* ws_ctx  = ws_qk + (size_t)B * H;

  const dim3 gblk(128, 1, 1);
  const dim3 ggrid(B / 16, H / 64, 1);

  // 1) LSTM cell -> h_new, c_new
  lstm_cell_wmma<<<ggrid, gblk, 0, stream>>>(pose, h, c, W_ih, W_hh,
                                             b_ih, b_hh, out_h, out_c, H);
  // 2) Q = query @ Wq^T + bq   (temp in attended slot)
  gemm_wmma<true><<<ggrid, gblk, 0, stream>>>(query, Wq, bq, out_att, H, H);
  // 3) Qk = Q @ Wk
  gemm_wmma<false><<<ggrid, gblk, 0, stream>>>(out_att, Wk, nullptr, ws_qk, H, H);
  // 4) single-pass attention: scores/softmax/ctx from a 256KB LDS stage
  attention_lds<<<dim3(B, 1, 1), dim3(256, 1, 1), 0, stream>>>(
      memory, out_h, ws_qk, ws_ctx);
  // 5) attended = ctx @ Wv^T + bv
  gemm_wmma<true><<<ggrid, gblk, 0, stream>>>(ws_ctx, Wv, bv, out_att, H, H);
}